// SplineRenderNet_19679540151016
// MI455X (gfx1250) — compile-verified
//
#include <hip/hip_runtime.h>
#include <hip/hip_bf16.h>

// SplineRender: B quadratic Beziers -> B x 224 x 224 distance-field images.
// Compute-bound (cubic root solve + transcendentals per pixel). No matmul
// structure => no WMMA; optimized for the VALU/TRANS pipes + NT b128 stores.

#define IMSIZE 224
#define NPIX   (IMSIZE * IMSIZE)   // 50176 = 49 * 1024
#define PIX_PER_BLOCK 1024         // 256 threads * 4 pixels

static constexpr float EPS_F = 1e-7f;
static constexpr float BIG_F = 1e30f;
static constexpr float LOG2E = 1.4426950408889634f;

typedef float v4f __attribute__((ext_vector_type(4)));

// cbrt(t) = sign(t) * max(|t|,1e-30)^(1/3) via v_log_f32 / v_exp_f32
__device__ __forceinline__ float fcbrt(float t) {
    float at = fmaxf(fabsf(t), 1e-30f);
    float r  = __builtin_amdgcn_exp2f(__builtin_amdgcn_logf(at) * (1.0f / 3.0f));
    return copysignf(r, t);
}

// dist2 = min(dist2, ok ? |P(t)-pix|^2 : BIG), P(t) = P0 + t*(2A + t*B)
__device__ __forceinline__ void accum_root(float& dist2, float r, bool v,
                                           float pos_x, float pos_y,
                                           float TAx, float TAy,
                                           float Bx, float By) {
    bool ok  = v && (r >= 0.0f) && (r <= 1.0f);
    float t  = fminf(fmaxf(r, 0.0f), 1.0f);
    float ex = fmaf(t, fmaf(t, Bx, TAx), pos_x);   // px - xp (sign-flipped, squared anyway)
    float ey = fmaf(t, fmaf(t, By, TAy), pos_y);
    float dd = fmaf(ex, ex, ey * ey);
    dist2 = fminf(dist2, ok ? dd : BIG_F);
}

__global__ __launch_bounds__(256) void spline_render_kernel(
    const float* __restrict__ gx0, const float* __restrict__ gy0,
    const float* __restrict__ gx1, const float* __restrict__ gy1,
    const float* __restrict__ gx2, const float* __restrict__ gy2,
    float* __restrict__ out)
{
    const int s  = blockIdx.y;                               // spline id (uniform)
    const int p0 = blockIdx.x * PIX_PER_BLOCK + threadIdx.x * 4;

    // Uniform per-block spline parameters -> scalar loads (s_load) by codegen.
    const float X0 = gx0[s], Y0 = gy0[s];
    const float X1 = gx1[s], Y1 = gy1[s];
    const float X2 = gx2[s], Y2 = gy2[s];

    const float Ax = X1 - X0,              Ay = Y1 - Y0;
    const float Bx = X0 - 2.0f*X1 + X2,    By = Y0 - 2.0f*Y1 + Y2;
    const float TAx = 2.0f*Ax,             TAy = 2.0f*Ay;
    const float a  = Bx*Bx + By*By;                          // uniform
    const float b  = 3.0f*(Ax*Bx + Ay*By);                   // uniform
    const float c0 = 2.0f*(Ax*Ax + Ay*Ay);                   // uniform
    const float dX20 = X2 - X0;

    // 4 consecutive pixels per thread; 224 % 4 == 0 so all share one row.
    const int   yi = p0 / IMSIZE;
    const float yp = (float)yi;
    const float xb = (float)(p0 - yi * IMSIZE);
    const float pos_y = Y0 - yp;
    const float p2y   = Y2 - yp;

    float res[4];

    if (fabsf(a) > EPS_F) {
        // ---- cubic path (wave-uniform branch) ----
        const float rcp_a = __builtin_amdgcn_rcpf(a);
        const float aa  = b * rcp_a;                 // uniform
        const float off = -aa * (1.0f / 3.0f);       // -aa/3
        const float k0  = (2.0f / 27.0f) * aa*aa*aa; // q = k0 + off*bb + cc
        const float paa = off * aa;                  // p = bb + paa
        #pragma unroll
        for (int j = 0; j < 4; ++j) {
            const float xp    = xb + (float)j;
            const float pos_x = X0 - xp;
            const float p2x   = pos_x + dX20;
            float dist2 = fminf(fmaf(pos_x, pos_x, pos_y*pos_y),
                                fmaf(p2x,  p2x,  p2y*p2y));

            const float cq = c0 + pos_x*Bx + pos_y*By;
            const float dq = pos_x*Ax + pos_y*Ay;
            const float bb = cq * rcp_a;
            const float cc = dq * rcp_a;
            const float p  = bb + paa;
            const float q  = fmaf(off, bb, k0) + cc;
            const float p3 = p * p * p;
            const float D  = fmaf(q, q, (4.0f / 27.0f) * p3);

            // D > eps : one real root via cbrt pair
            const float sqD   = __builtin_amdgcn_sqrtf(fmaxf(D, 0.0f));
            const float h     = -0.5f * q;
            const float r_pos = fcbrt(h + 0.5f*sqD) + fcbrt(h - 0.5f*sqD) + off;

            // D < -eps : three real roots via trig; one cos + sqrt, angle addition
            const bool  three  = D < -EPS_F;
            const float p_neg  = three ? p  : -1.0f;
            const float p3_neg = three ? p3 : -1.0f;
            const float u   = 2.0f * __builtin_amdgcn_sqrtf(-p_neg * (1.0f / 3.0f));
            float arg = -2.598076211353316f * q * __builtin_amdgcn_rsqf(-p3_neg); // -sqrt(27)/2 * q / sqrt(-p3)
            arg = fminf(fmaxf(arg, -1.0f), 1.0f);
            const float phi = acosf(arg) * (1.0f / 3.0f);
            const float cph = __cosf(phi);                                  // phi in [0, pi/3] -> sin >= 0
            const float sph = __builtin_amdgcn_sqrtf(fmaxf(fmaf(-cph, cph, 1.0f), 0.0f));
            const float rt0 = fmaf(u, cph, off);
            const float rt1 = fmaf(u, fmaf(-0.8660254037844386f, sph, -0.5f*cph), off); // cos(phi+2pi/3)
            const float rt2 = fmaf(u, fmaf( 0.8660254037844386f, sph, -0.5f*cph), off); // cos(phi+4pi/3)

            // |D| <= eps : repeated root
            const float u0  = fcbrt(h);           // cbrt(-q/2)
            const float rn1 = fmaf(2.0f, u0, off);
            const float rn2 = off - u0;

            const float r1 = (D > EPS_F) ? r_pos : (three ? rt0 : rn1);
            const float r2 = three ? rt1 : rn2;
            const bool  v2 = (D <= EPS_F);

            accum_root(dist2, r1,  true,  pos_x, pos_y, TAx, TAy, Bx, By);
            accum_root(dist2, r2,  v2,    pos_x, pos_y, TAx, TAy, Bx, By);
            accum_root(dist2, rt2, three, pos_x, pos_y, TAx, TAy, Bx, By);

            res[j] = __builtin_amdgcn_exp2f(-LOG2E * dist2);   // exp(-dist2/FUZZ), FUZZ=1
        }
    } else if (fabsf(b) > EPS_F) {
        // ---- quadratic fallback (wave-uniform) ----
        const float inv2a = __builtin_amdgcn_rcpf(2.0f * b);
        #pragma unroll
        for (int j = 0; j < 4; ++j) {
            const float xp    = xb + (float)j;
            const float pos_x = X0 - xp;
            const float p2x   = pos_x + dX20;
            float dist2 = fminf(fmaf(pos_x, pos_x, pos_y*pos_y),
                                fmaf(p2x,  p2x,  p2y*p2y));

            const float cq = c0 + pos_x*Bx + pos_y*By;
            const float dq = pos_x*Ax + pos_y*Ay;
            const float Dq = fmaf(cq, cq, -4.0f * b * dq);
            const float sq = __builtin_amdgcn_sqrtf(fmaxf(Dq, 0.0f));
            const float r1 = ((Dq > EPS_F) ? (-cq - sq) : -cq) * inv2a;
            const bool  v1 = (Dq >= -EPS_F);
            const float r2 = (-cq + sq) * inv2a;
            const bool  v2 = (Dq > EPS_F);

            accum_root(dist2, r1, v1, pos_x, pos_y, TAx, TAy, Bx, By);
            accum_root(dist2, r2, v2, pos_x, pos_y, TAx, TAy, Bx, By);

            res[j] = __builtin_amdgcn_exp2f(-LOG2E * dist2);
        }
    } else {
        // ---- linear fallback (wave-uniform) ----
        #pragma unroll
        for (int j = 0; j < 4; ++j) {
            const float xp    = xb + (float)j;
            const float pos_x = X0 - xp;
            const float p2x   = pos_x + dX20;
            float dist2 = fminf(fmaf(pos_x, pos_x, pos_y*pos_y),
                                fmaf(p2x,  p2x,  p2y*p2y));

            const float cq = c0 + pos_x*Bx + pos_y*By;
            const float dq = pos_x*Ax + pos_y*Ay;
            const bool  v1 = fabsf(cq) > EPS_F;
            const float r1 = -dq * __builtin_amdgcn_rcpf(v1 ? cq : 1.0f);

            accum_root(dist2, r1, v1, pos_x, pos_y, TAx, TAy, Bx, By);

            res[j] = __builtin_amdgcn_exp2f(-LOG2E * dist2);
        }
    }

    // 128-bit non-temporal store (write-once stream, don't pollute L2).
    v4f v = { res[0], res[1], res[2], res[3] };
    __builtin_nontemporal_store(v, (v4f*)(out + (size_t)s * NPIX + p0));
}

extern "C" void kernel_launch(void* const* d_in, const int* in_sizes, int n_in,
                              void* d_out, int out_size, void* d_ws, size_t ws_size,
                              hipStream_t stream) {
    const float* x0 = (const float*)d_in[0];
    const float* y0 = (const float*)d_in[1];
    const float* x1 = (const float*)d_in[2];
    const float* y1 = (const float*)d_in[3];
    const float* x2 = (const float*)d_in[4];
    const float* y2 = (const float*)d_in[5];
    float* out = (float*)d_out;

    const int B = in_sizes[0];                      // 256 splines
    dim3 grid(NPIX / PIX_PER_BLOCK, B);             // 49 x B blocks
    spline_render_kernel<<<grid, 256, 0, stream>>>(x0, y0, x1, y1, x2, y2, out);
}